// ImageTranslationLayer_1975684956223
// MI455X (gfx1250) — compile-verified
//
#include <hip/hip_runtime.h>
#include <math.h>

// Problem constants (from the reference): image [32,1024,1024,1], max_shift=20.
#define Z_DIM 32
#define H_DIM 1024
#define W_DIM 1024
#define OFF   10                    // max_shift // 2
#define OH    (H_DIM - 2 * OFF)     // 1004
#define OW    (W_DIM - 2 * OFF)     // 1004
#define SEG   1008                  // staged columns per row (1004 + clamp margin)
#define NTHREADS 256                // 8 wave32 waves

#if __has_builtin(__builtin_amdgcn_global_load_async_to_lds_b32)
#define HAVE_ASYNC_LDS 1
#else
#define HAVE_ASYNC_LDS 0
#warning "gfx1250 __builtin_amdgcn_global_load_async_to_lds_b32 NOT available; using plain LDS staging fallback"
#endif

// The builtin's params are (as(1) int*, as(3) int*, imm offset, imm cpol).
typedef __attribute__((address_space(1))) int* g32p;
typedef __attribute__((address_space(3))) int* l32p;

__launch_bounds__(NTHREADS)
__global__ void translate_crop_kernel(const float* __restrict__ img,
                                      const float* __restrict__ trans,
                                      float* __restrict__ out)
{
    __shared__ float smem[2][SEG];   // two input-row segments, ~8 KB of 320 KB LDS

    const int z = blockIdx.y;        // slice
    const int i = blockIdx.x;        // output row in [0, OH)

    // Per-slice pixel shift. Reference: theta x-row offset = t0/(H/2),
    // y-row offset = t1/(W/2); mapped to pixels -> dx = t0*(W-1)/H, dy = t1*(H-1)/W.
    const float t0 = trans[2 * z + 0];
    const float t1 = trans[2 * z + 1];
    const float dx = t0 * ((float)(W_DIM - 1) / (float)H_DIM);
    const float dy = t1 * ((float)(H_DIM - 1) / (float)W_DIM);

    // Row coordinates (uniform per block -> SALU-friendly).
    const float y   = (float)(i + OFF) + dy;
    const float y0f = floorf(y);
    int y0 = (int)y0f;     y0 = min(max(y0, 0), H_DIM - 1);
    int y1 = (int)y0f + 1; y1 = min(max(y1, 0), H_DIM - 1);
    const float wyA = (float)y1 - y;   // weight on row y0 (reference: y1c - y)
    const float wyB = y - (float)y0;   // weight on row y1 (reference: y - y0c)

    // Column window start, with 1-column guard for fp rounding of floor((j+10)+dx).
    int lo = (int)floorf((float)OFF + dx) - 1;
    lo = min(max(lo, 0), W_DIM - 1);

    const float* row0 = img + ((size_t)z * H_DIM + y0) * W_DIM;
    const float* row1 = img + ((size_t)z * H_DIM + y1) * W_DIM;

    // ---- Stage: async global -> LDS copy of both row segments (ASYNCcnt path) ----
    const int tid = threadIdx.x;
    for (int c = tid; c < SEG; c += NTHREADS) {
        const int gcol = min(lo + c, W_DIM - 1);   // clamp tail duplicates; harmless
#if HAVE_ASYNC_LDS
        // strip const with a generic-pointer cast, then addrspacecast to as(1)/as(3)
        __builtin_amdgcn_global_load_async_to_lds_b32(
            (g32p)(float*)(row0 + gcol), (l32p)&smem[0][c], 0, 0);
        __builtin_amdgcn_global_load_async_to_lds_b32(
            (g32p)(float*)(row1 + gcol), (l32p)&smem[1][c], 0, 0);
#else
        smem[0][c] = row0[gcol];
        smem[1][c] = row1[gcol];
#endif
    }
#if HAVE_ASYNC_LDS
#if __has_builtin(__builtin_amdgcn_s_wait_asynccnt)
    __builtin_amdgcn_s_wait_asynccnt(0);
#else
    asm volatile("s_wait_asynccnt 0x0" ::: "memory");
#endif
#endif
    __syncthreads();

    // ---- Compute: 4-tap blend from LDS, NT stores for the write-once output ----
    float* orow = out + ((size_t)z * OH + i) * OW;
    for (int j = tid; j < OW; j += NTHREADS) {
        const float x   = (float)(j + OFF) + dx;
        const float x0f = floorf(x);
        int x0 = (int)x0f;     x0 = min(max(x0, 0), W_DIM - 1);
        int x1 = (int)x0f + 1; x1 = min(max(x1, 0), W_DIM - 1);
        const float wxA = (float)x1 - x;   // reference: x1c - x
        const float wxB = x - (float)x0;   // reference: x - x0c

        const int l0 = min(max(x0 - lo, 0), SEG - 1);
        const int l1 = min(max(x1 - lo, 0), SEG - 1);

        const float Ia = smem[0][l0];   // (y0, x0)
        const float Ib = smem[1][l0];   // (y1, x0)
        const float Ic = smem[0][l1];   // (y0, x1)
        const float Id = smem[1][l1];   // (y1, x1)

        const float v = (wxA * wyA) * Ia + (wxA * wyB) * Ib +
                        (wxB * wyA) * Ic + (wxB * wyB) * Id;
        __builtin_nontemporal_store(v, orow + j);
    }
}

extern "C" void kernel_launch(void* const* d_in, const int* in_sizes, int n_in,
                              void* d_out, int out_size, void* d_ws, size_t ws_size,
                              hipStream_t stream)
{
    const float* img   = (const float*)d_in[0];   // [32,1024,1024,1] f32
    const float* trans = (const float*)d_in[1];   // [32,2,1] f32
    // d_in[2] = max_shift scalar (constant 20 per reference; OFF hardcoded)
    float* out = (float*)d_out;                   // [32,1004,1004,1] f32

    dim3 grid(OH, Z_DIM);
    translate_crop_kernel<<<grid, NTHREADS, 0, stream>>>(img, trans, out);
}